// CapsuleNetwork_27900107555003
// MI455X (gfx1250) — compile-verified
//
#include <hip/hip_runtime.h>
#include <math.h>

typedef __attribute__((ext_vector_type(2))) float v2f;
typedef __attribute__((ext_vector_type(8))) float v8f;

#define B_TOT 128
#define N_TOT 1152
#define D_TOT 45
#define O_TOT 16
#define P_TOT 8
#define BT    16   // batch tile (one WMMA N-tile)
#define NT    16   // capsules-in per block
#define DMAX  6    // ceil(45/8) d-tiles per wave

// ---------------- zero scratch ----------------
__global__ __launch_bounds__(256) void caps_zero(float* __restrict__ p, int n) {
    int i = blockIdx.x * blockDim.x + threadIdx.x;
    if (i < n) p[i] = 0.0f;
}

// ---------------- fused routing pass ----------------
// For each (b-tile, n-chunk): recompute u_hat tiles with v_wmma_f32_16x16x4_f32,
// logits = sum_o u_hat * (va+vb)   (linearity: b_i = sum of past agreements),
// c = softmax_d(logits), accumulate s += c * u_hat, atomic-add into global s.
// va == nullptr => iteration 0 (uniform c = 1/45, no softmax needed).
__global__ __launch_bounds__(256) void caps_route_pass(
    const float* __restrict__ u,   // (B, N, P)
    const float* __restrict__ W,   // (N, D, O, P)
    const float* __restrict__ va,  // (B, D, O) or null
    const float* __restrict__ vb,  // (B, D, O) or null
    float* __restrict__ s_out)     // (B, D, O), pre-zeroed
{
    __shared__ float sh_agr[D_TOT * BT];
    __shared__ float sh_c[D_TOT * BT];

    const int tid  = threadIdx.x;
    const int lane = tid & 31;
    const int wave = tid >> 5;       // 8 waves
    const int lo   = lane & 15;      // tile column (batch) / A-row (o)
    const int hi   = lane >> 4;      // half-wave select
    const int n0   = blockIdx.x * NT;
    const int b0   = blockIdx.y * BT;
    const int bg   = b0 + lo;        // this lane's batch for B/C fragments

    const bool has_v = (va != nullptr);

    // v fragments in C layout: vs[j][r] = (va+vb)[bg, d, o = r + 8*hi]
    float vs[DMAX][8];
    #pragma unroll
    for (int j = 0; j < DMAX; ++j) {
        int d = wave + 8 * j;
        bool ok = (d < D_TOT);
        #pragma unroll
        for (int r = 0; r < 8; ++r) {
            int o = r + 8 * hi;
            int idx = (bg * D_TOT + (ok ? d : 0)) * O_TOT + o;
            float x = (has_v && ok) ? va[idx] : 0.0f;
            if (vb && ok) x += vb[idx];
            vs[j][r] = x;
        }
    }

    v8f sacc[DMAX];
    #pragma unroll
    for (int j = 0; j < DMAX; ++j) sacc[j] = (v8f){};

    for (int nn = 0; nn < NT; ++nn) {
        const int n = n0 + nn;
        if (nn + 1 < NT)  // pull next capsule's weight slab toward the WGP
            __builtin_prefetch(&W[(n + 1) * D_TOT * O_TOT * P_TOT], 0, 1);

        // B fragment (shared by every d-tile): rows K, cols = batches.
        // bf[kk]: lanes 0-15 rows {4kk+0,4kk+1}, lanes 16-31 rows {4kk+2,4kk+3}.
        v2f bf[2];
        {
            const float* up = &u[(bg * N_TOT + n) * P_TOT];
            #pragma unroll
            for (int kk = 0; kk < 2; ++kk) {
                int p = kk * 4 + 2 * hi;
                bf[kk].x = up[p];
                bf[kk].y = up[p + 1];
            }
        }

        // u_hat tiles: A = W[n,d,:,:] (16 o x 8 p), C[o, b] f32
        v8f uh[DMAX];
        #pragma unroll
        for (int j = 0; j < DMAX; ++j) {
            int d = wave + 8 * j;   // wave-uniform -> EXEC stays all-ones
            if (d < D_TOT) {
                const float* wp = &W[((n * D_TOT + d) * O_TOT + lo) * P_TOT];
                v8f acc = (v8f){};
                #pragma unroll
                for (int kk = 0; kk < 2; ++kk) {
                    int p = kk * 4 + 2 * hi;
                    v2f af;
                    af.x = wp[p];
                    af.y = wp[p + 1];
                    acc = __builtin_amdgcn_wmma_f32_16x16x4_f32(
                        false, af, false, bf[kk], (short)0, acc, false, false);
                }
                uh[j] = acc;
            } else {
                uh[j] = (v8f){};
            }
        }

        // routing logits: agr[b,d] = sum_o u_hat[o,b] * vs[o,b]
        if (has_v) {
            #pragma unroll
            for (int j = 0; j < DMAX; ++j) {
                int d = wave + 8 * j;
                float ap = 0.0f;
                #pragma unroll
                for (int r = 0; r < 8; ++r) ap += uh[j][r] * vs[j][r];
                float full = ap + __shfl_xor(ap, 16, 32);  // join o halves
                if (d < D_TOT && hi == 0) sh_agr[d * BT + lo] = full;
            }
        }
        __syncthreads();

        // softmax over d (45 values) per (batch, n)
        if (has_v && tid < BT) {
            float m = -3.4e38f;
            for (int d = 0; d < D_TOT; ++d)
                m = fmaxf(m, sh_agr[d * BT + tid]);
            float sum = 0.0f;
            for (int d = 0; d < D_TOT; ++d) {
                float e = __expf(sh_agr[d * BT + tid] - m);
                sh_c[d * BT + tid] = e;
                sum += e;
            }
            float inv = 1.0f / sum;
            for (int d = 0; d < D_TOT; ++d) sh_c[d * BT + tid] *= inv;
        }
        __syncthreads();

        // s += c * u_hat
        #pragma unroll
        for (int j = 0; j < DMAX; ++j) {
            int d = wave + 8 * j;
            if (d < D_TOT) {
                float cj = has_v ? sh_c[d * BT + lo] : (1.0f / 45.0f);
                #pragma unroll
                for (int r = 0; r < 8; ++r) sacc[j][r] += cj * uh[j][r];
            }
        }
        __syncthreads();
    }

    // reduce partial s across n-chunks
    #pragma unroll
    for (int j = 0; j < DMAX; ++j) {
        int d = wave + 8 * j;
        if (d < D_TOT) {
            #pragma unroll
            for (int r = 0; r < 8; ++r) {
                int o = r + 8 * hi;
                atomicAdd(&s_out[(bg * D_TOT + d) * O_TOT + o], sacc[j][r]);
            }
        }
    }
}

// ---------------- squash ----------------
__global__ __launch_bounds__(256) void caps_squash(
    const float* __restrict__ s, float* __restrict__ v, int cnt)
{
    int i = blockIdx.x * blockDim.x + threadIdx.x;
    if (i >= cnt) return;  // cnt = B*D
    const float* sp = &s[i * O_TOT];
    float x[O_TOT];
    float ns = 0.0f;
    #pragma unroll
    for (int r = 0; r < O_TOT; ++r) { x[r] = sp[r]; ns += x[r] * x[r]; }
    float norm = sqrtf(ns);
    float sc = ns / (1.0f + ns) / (norm + 1e-8f);
    #pragma unroll
    for (int r = 0; r < O_TOT; ++r) v[i * O_TOT + r] = sc * x[r];
}

extern "C" void kernel_launch(void* const* d_in, const int* in_sizes, int n_in,
                              void* d_out, int out_size, void* d_ws, size_t ws_size,
                              hipStream_t stream) {
    const float* u = (const float*)d_in[0];   // (128, 1152, 8)
    const float* W = (const float*)d_in[1];   // (1, 1152, 45, 16, 8)
    float* out = (float*)d_out;               // (128, 45, 16)

    const int SVO = B_TOT * D_TOT * O_TOT;    // 92160
    float* s  = (float*)d_ws;
    float* v0 = s + SVO;
    float* v1 = v0 + SVO;

    dim3 pgrid(N_TOT / NT, B_TOT / BT);       // (72, 8)
    const int zb = (SVO + 255) / 256;
    const int qb = (B_TOT * D_TOT + 255) / 256;

    // iter 0: uniform c
    caps_zero<<<zb, 256, 0, stream>>>(s, SVO);
    caps_route_pass<<<pgrid, 256, 0, stream>>>(u, W, nullptr, nullptr, s);
    caps_squash<<<qb, 256, 0, stream>>>(s, v0, B_TOT * D_TOT);

    // iter 1: logits = agr(v0)
    caps_zero<<<zb, 256, 0, stream>>>(s, SVO);
    caps_route_pass<<<pgrid, 256, 0, stream>>>(u, W, v0, nullptr, s);
    caps_squash<<<qb, 256, 0, stream>>>(s, v1, B_TOT * D_TOT);

    // iter 2: logits = agr(v0) + agr(v1)
    caps_zero<<<zb, 256, 0, stream>>>(s, SVO);
    caps_route_pass<<<pgrid, 256, 0, stream>>>(u, W, v0, v1, s);
    caps_squash<<<qb, 256, 0, stream>>>(s, out, B_TOT * D_TOT);
}